// RecurrentMemoryAttention_17514876634075
// MI455X (gfx1250) — compile-verified
//
#include <hip/hip_runtime.h>
#include <hip/hip_bf16.h>

#define B_SZ    512
#define DMODEL  1024
#define DHEAD   64
#define NHEADS  64
#define K_ACT   8
#define EPSF    1e-6f
#define TM      16
#define KC      64          // K-chunk staged per TDM transfer

typedef __attribute__((ext_vector_type(2))) float v2f;
typedef __attribute__((ext_vector_type(8))) float v8f;
typedef __attribute__((ext_vector_type(4))) unsigned int v4u;
typedef __attribute__((ext_vector_type(8))) int v8i;
typedef __attribute__((ext_vector_type(4))) int v4i;

__device__ __forceinline__ v8f wmma4(v2f a, v2f b, v8f c) {
  // V_WMMA_F32_16X16X4_F32 : D = A(16x4) * B(4x16) + C(16x16), fp32 throughout
  return __builtin_amdgcn_wmma_f32_16x16x4_f32(false, a, false, b, (short)0, c,
                                               false, false);
}

// TDM: DMA a [kc x 64] fp32 tile (row stride 64) from global into LDS.
// D# group0: count=1 | lds_addr | global_addr | type=2 ("image")
// D# group1: data_size=4B, tensor_dim0=64, tensor_dim1=kc, tile=64 x kc, stride0=64
__device__ __forceinline__ void tdm_load_w(const float* gsrc, float* lds_dst,
                                           int kc_rows) {
  unsigned long long ga = (unsigned long long)(uintptr_t)gsrc;
  unsigned int la = (unsigned int)(uintptr_t)lds_dst;   // LDS byte offset
  v4u g0;
  g0.x = 1u;                                            // count=1, user D#
  g0.y = la;                                            // lds_addr [63:32]
  g0.z = (unsigned int)(ga & 0xFFFFFFFFu);              // global_addr lo
  g0.w = (unsigned int)((ga >> 32) & 0x01FFFFFFu)       // global_addr hi
         | (2u << 30);                                  // type=2
  v8i g1;
  g1[0] = 0x00020000;                 // workgroup_mask=0, data_size=2 (4 bytes)
  g1[1] = (DHEAD << 16);              // tensor_dim0 = 64 (low16 @ bits63:48)
  g1[2] = (kc_rows & 0xFFFF) << 16;   // dim0 hi=0 ; tensor_dim1 low16
  g1[3] = (DHEAD << 16);              // dim1 hi=0 ; tile_dim0 = 64
  g1[4] = kc_rows & 0xFFFF;           // tile_dim1 = kc ; tile_dim2 = 0
  g1[5] = DHEAD;                      // tensor_dim0_stride = 64
  g1[6] = 0;
  g1[7] = 0;
  v4i z4 = {0, 0, 0, 0};
  v8i z8 = {0, 0, 0, 0, 0, 0, 0, 0};
  // clang-23 / therock form: (g0, g1, g2, g3, g4, cpol)
  __builtin_amdgcn_tensor_load_to_lds(g0, g1, z4, z4, z8, 0);
}

// ---------------- kernel 0: zero output + head counters ----------------
__global__ void zero_kernel(float* __restrict__ out, int n, int* __restrict__ head_count) {
  int i = blockIdx.x * blockDim.x + threadIdx.x;
  if (i < n) out[i] = 0.0f;
  if (i < NHEADS) head_count[i] = 0;
}

// ---------------- kernel 1: routing (softmax + top-8 + row lists) ----------------
__global__ __launch_bounds__(64)
void route_kernel(const float* __restrict__ q,
                  const float* __restrict__ rw,
                  const float* __restrict__ rb,
                  float* __restrict__ probs_ws,
                  int* __restrict__ sel_ws,
                  int* __restrict__ head_count,
                  int* __restrict__ head_rows) {
  __shared__ float pf[NHEADS];
  __shared__ float rv[NHEADS];
  __shared__ int   ri[NHEADS];
  __shared__ float topv[K_ACT];
  __shared__ int   topi[K_ACT];
  __shared__ float rnorm;

  const int b = blockIdx.x;
  const int t = threadIdx.x;                     // 0..63 : one head per thread
  const float* qr = q + (size_t)b * DMODEL;

  float acc = rb[t];
  for (int d = 0; d < DMODEL; ++d)
    acc += qr[d] * rw[d * NHEADS + t];

  // softmax: max reduce
  rv[t] = acc; __syncthreads();
  for (int s = 32; s > 0; s >>= 1) {
    if (t < s) rv[t] = fmaxf(rv[t], rv[t + s]);
    __syncthreads();
  }
  float m = rv[0]; __syncthreads();
  float e = __expf(acc - m);
  rv[t] = e; __syncthreads();
  for (int s = 32; s > 0; s >>= 1) {
    if (t < s) rv[t] += rv[t + s];
    __syncthreads();
  }
  float inv = 1.0f / rv[0];
  __syncthreads();
  pf[t] = e * inv;
  __syncthreads();

  // top-8 by repeated argmax
  for (int it = 0; it < K_ACT; ++it) {
    rv[t] = pf[t]; ri[t] = t; __syncthreads();
    for (int s = 32; s > 0; s >>= 1) {
      if (t < s && rv[t + s] > rv[t]) { rv[t] = rv[t + s]; ri[t] = ri[t + s]; }
      __syncthreads();
    }
    if (t == 0) { topv[it] = rv[0]; topi[it] = ri[0]; pf[ri[0]] = -1.0f; }
    __syncthreads();
  }

  if (t == 0) {
    float s = 0.0f;
    for (int i = 0; i < K_ACT; ++i) s += topv[i];
    rnorm = 1.0f / (s + EPSF);
  }
  __syncthreads();

  if (t < K_ACT) {
    int h = topi[t];
    float p = topv[t] * rnorm;
    sel_ws[b * K_ACT + t] = h;
    probs_ws[b * K_ACT + t] = p;
    int pos = atomicAdd(&head_count[h], 1);
    head_rows[h * B_SZ + pos] = (b << 3) | t;   // pack (batch row, slot)
  }
}

// ---------------- kernel 2: per-head QKV + memory read + combiner ----------------
__global__ __launch_bounds__(128)
void head_kernel(const float* __restrict__ queries,
                 const float* __restrict__ mem_matrix,
                 const float* __restrict__ mem_norm,
                 const float* __restrict__ wq, const float* __restrict__ bq,
                 const float* __restrict__ wk, const float* __restrict__ bk,
                 const float* __restrict__ wv, const float* __restrict__ bv,
                 const float* __restrict__ wc, const float* __restrict__ bc,
                 const float* __restrict__ probs_ws,
                 const int* __restrict__ head_count,
                 const int* __restrict__ head_rows,
                 float* __restrict__ out) {
  __shared__ int   sb_b[TM];
  __shared__ float sb_p[TM];
  __shared__ float q_s[TM * DHEAD];
  __shared__ float k_s[TM * DHEAD];
  __shared__ float v_s[TM * DHEAD];
  __shared__ float a_s[TM * DHEAD];
  extern __shared__ float stage[];    // [2 bufs][3 mats][KC * DHEAD] = 96 KB

  const int h     = blockIdx.x;
  const int nrows = head_count[h];
  const int tid   = threadIdx.x;
  const int lane  = tid & 31;
  const int wave  = tid >> 5;           // 0..3
  const int lrow  = lane & 15;          // row (A frag) / col (B and C/D frag) in tile
  const int khalf = (lane >> 4) * 2;    // K offset: lanes 16-31 hold K+2,K+3
  const int hi8   = (lane >> 4) * 8;    // C/D: lanes 16-31 hold rows 8..15

  const float* Wq = wq + (size_t)h * DMODEL * DHEAD;
  const float* Wk = wk + (size_t)h * DMODEL * DHEAD;
  const float* Wv = wv + (size_t)h * DMODEL * DHEAD;
  const float* Wc = wc + (size_t)h * DHEAD * DMODEL;
  const float* Bq = bq + h * DHEAD;
  const float* Bk = bk + h * DHEAD;
  const float* Bv = bv + h * DHEAD;
  const float* Bc = bc + h * DMODEL;

  const int CHUNK = KC * DHEAD;         // floats per staged matrix chunk

  for (int t0 = (int)blockIdx.y * TM; t0 < nrows; t0 += (int)gridDim.y * TM) {
    if (tid < TM) {
      int idx = t0 + tid;
      int packed = head_rows[h * B_SZ + (idx < nrows ? idx : t0)];
      sb_b[tid] = packed >> 3;
      sb_p[tid] = probs_ws[(packed >> 3) * K_ACT + (packed & 7)];
    }
    __syncthreads();

    // ---- Phase 1: Q/K/V = X_tile[16x1024] @ W[1024x64] ----
    // TDM double-buffers W chunks into LDS; WMMA consumes from LDS.
    {
      const int nb  = wave * 16;                      // each wave owns one N-tile
      const int col = nb + lrow;
      v8f accq = {}; v8f acck = {}; v8f accv = {};
      const float* qrow = queries + (size_t)sb_b[lrow] * DMODEL;

      if (wave == 0) {                                // prologue: DMA chunk 0
        tdm_load_w(Wq, stage,             KC);
        tdm_load_w(Wk, stage + CHUNK,     KC);
        tdm_load_w(Wv, stage + 2 * CHUNK, KC);
      }

      for (int ck = 0; ck < DMODEL / KC; ++ck) {
        if (wave == 0) {
          if (ck + 1 < DMODEL / KC) {                 // DMA next chunk, other buf
            float* nx = stage + ((ck + 1) & 1) * 3 * CHUNK;
            const size_t off = (size_t)(ck + 1) * KC * DHEAD;
            tdm_load_w(Wq + off, nx,             KC);
            tdm_load_w(Wk + off, nx + CHUNK,     KC);
            tdm_load_w(Wv + off, nx + 2 * CHUNK, KC);
            __builtin_amdgcn_s_wait_tensorcnt(3);     // chunk ck done (in-order)
          } else {
            __builtin_amdgcn_s_wait_tensorcnt(0);
          }
        }
        __syncthreads();                              // stage[ck&1] ready

        const float* sq = stage + (ck & 1) * 3 * CHUNK;
        const float* sk = sq + CHUNK;
        const float* sv = sq + 2 * CHUNK;
        const int kbase = ck * KC;
        for (int kk = 0; kk < KC; kk += 4) {
          v2f a;
          a.x = qrow[kbase + kk + khalf];
          a.y = qrow[kbase + kk + khalf + 1];
          int r0 = (kk + khalf) * DHEAD + col;
          v2f fq, fk, fv;
          fq.x = sq[r0]; fq.y = sq[r0 + DHEAD];
          fk.x = sk[r0]; fk.y = sk[r0 + DHEAD];
          fv.x = sv[r0]; fv.y = sv[r0 + DHEAD];
          accq = wmma4(a, fq, accq);
          acck = wmma4(a, fk, acck);
          accv = wmma4(a, fv, accv);
        }
        __syncthreads();                              // all consumed before next DMA
      }

      #pragma unroll
      for (int j = 0; j < 8; ++j) {
        int r = j + hi8;
        float qv = accq[j] + Bq[col];
        float kv = acck[j] + Bk[col];
        q_s[r * DHEAD + col] = qv > 0.0f ? qv + 1.0f : __expf(qv);  // elu+1
        k_s[r * DHEAD + col] = kv > 0.0f ? kv + 1.0f : __expf(kv);
        v_s[r * DHEAD + col] = accv[j] + Bv[col];
      }
    }
    __syncthreads();

    // ---- Phase 2: memory read with rank-1 correction:
    //      attn = (q@M + (q.k)*v) / (q.(norm + k) + eps) ----
    {
      const int r   = tid >> 3;          // 16 rows
      const int sub = tid & 7;           // 8 threads x 8 cols per row
      const int b   = sb_b[r];
      const float* M   = mem_matrix + (((size_t)b * NHEADS + h) * DHEAD) * DHEAD;
      const float* Nrm = mem_norm   + ((size_t)b * NHEADS + h) * DHEAD;
      float den = 0.0f, qk = 0.0f;
      float num[8] = {0,0,0,0,0,0,0,0};
      for (int e = 0; e < DHEAD; ++e) {
        float qe = q_s[r * DHEAD + e];
        float ke = k_s[r * DHEAD + e];
        den += qe * (Nrm[e] + ke);
        qk  += qe * ke;
        const float* Mrow = M + (size_t)e * DHEAD + sub * 8;
        #pragma unroll
        for (int j = 0; j < 8; ++j) num[j] += qe * Mrow[j];
      }
      float rden = 1.0f / (den + EPSF);
      #pragma unroll
      for (int j = 0; j < 8; ++j) {
        int c = sub * 8 + j;
        a_s[r * DHEAD + c] = (num[j] + qk * v_s[r * DHEAD + c]) * rden;
      }
    }
    __syncthreads();

    // ---- Phase 3: combiner attn[16x64] @ Wc[64x1024], prob-weighted scatter ----
    for (int nt = wave; nt < DMODEL / 16; nt += 4) {
      const int nb  = nt * 16;
      const int col = nb + lrow;
      v8f acc = {};
      for (int k0 = 0; k0 < DHEAD; k0 += 4) {
        v2f a;
        a.x = a_s[lrow * DHEAD + k0 + khalf];
        a.y = a_s[lrow * DHEAD + k0 + khalf + 1];
        int r0 = (k0 + khalf) * DMODEL + col;
        v2f f;
        f.x = Wc[r0]; f.y = Wc[r0 + DMODEL];
        acc = wmma4(a, f, acc);
      }
      #pragma unroll
      for (int j = 0; j < 8; ++j) {
        int r = j + hi8;
        if (t0 + r < nrows) {
          float val = (acc[j] + Bc[col]) * sb_p[r];
          atomicAdd(out + (size_t)sb_b[r] * DMODEL + col, val);
        }
      }
    }
    __syncthreads();
  }
}

// ---------------- launcher ----------------
extern "C" void kernel_launch(void* const* d_in, const int* in_sizes, int n_in,
                              void* d_out, int out_size, void* d_ws, size_t ws_size,
                              hipStream_t stream) {
  const float* queries    = (const float*)d_in[0];
  const float* mem_matrix = (const float*)d_in[1];
  const float* mem_norm   = (const float*)d_in[2];
  const float* router_w   = (const float*)d_in[3];
  const float* router_b   = (const float*)d_in[4];
  const float* wq = (const float*)d_in[5];
  const float* bq = (const float*)d_in[6];
  const float* wk = (const float*)d_in[7];
  const float* bk = (const float*)d_in[8];
  const float* wv = (const float*)d_in[9];
  const float* bv = (const float*)d_in[10];
  const float* wc = (const float*)d_in[11];
  const float* bc = (const float*)d_in[12];
  float* out = (float*)d_out;

  char* ws = (char*)d_ws;
  float* probs_ws  = (float*)(ws);                          // B*K floats
  int*   sel_ws    = (int*)(ws + 4096 * 4);                 // B*K ints
  int*   head_count= (int*)(ws + 8192 * 4);                 // 64 ints
  int*   head_rows = (int*)(ws + 8192 * 4 + 256);           // 64*B ints

  int n = B_SZ * DMODEL;
  hipLaunchKernelGGL(zero_kernel, dim3((n + 255) / 256), dim3(256), 0, stream,
                     out, n, head_count);
  hipLaunchKernelGGL(route_kernel, dim3(B_SZ), dim3(64), 0, stream,
                     queries, router_w, router_b, probs_ws, sel_ws,
                     head_count, head_rows);
  size_t stage_bytes = 2ull * 3ull * KC * DHEAD * sizeof(float);  // 96 KB
  hipLaunchKernelGGL(head_kernel, dim3(NHEADS, 8), dim3(128), stage_bytes, stream,
                     queries, mem_matrix, mem_norm,
                     wq, bq, wk, bk, wv, bv, wc, bc,
                     probs_ws, head_count, head_rows, out);
}